// GINWithJK_38371237822822
// MI455X (gfx1250) — compile-verified
//
#include <hip/hip_runtime.h>
#include <hip/hip_bf16.h>

#define Nn 50000
#define Ee 800000
#define FIN 128
#define HH 64
#define LL 5
#define CC 10
#define GG 512

typedef __attribute__((ext_vector_type(2))) float v2f;
typedef __attribute__((ext_vector_type(8))) float v8f;

__global__ __launch_bounds__(256) void zero_f32(float* __restrict__ p, int n) {
    int i = blockIdx.x * blockDim.x + threadIdx.x;
    if (i < n) p[i] = 0.0f;
}

// one thread per (edge, feature); feature-major within an edge -> coalesced
__global__ __launch_bounds__(256) void scatter_add(const float* __restrict__ xin, int xstride,
                                                   const int* __restrict__ src,
                                                   const int* __restrict__ dst,
                                                   float* __restrict__ agg, int lf, int total) {
    int i = blockIdx.x * blockDim.x + threadIdx.x;
    if (i >= total) return;
    int e = i >> lf;
    int f = i & ((1 << lf) - 1);
    int s = src[e];
    int d = dst[e];
    atomicAdd(&agg[(d << lf) + f], xin[s * xstride + f]);
}

// One block = 16 nodes; 4 waves, wave w computes output columns [16w, 16w+16).
// GEMM1: h1 = relu(((1+eps)x + agg) @ wa + ba)   (K = F, fp32 WMMA 16x16x4)
// GEMM2: h2 = relu(h1 @ wb + bb); BN eval; write em slice.
__global__ __launch_bounds__(128) void gin_mlp(
    const float* __restrict__ xin, int xstride, int K,
    const float* __restrict__ agg,
    const float* __restrict__ epsp,
    const float* __restrict__ wa, const float* __restrict__ ba,
    const float* __restrict__ wb, const float* __restrict__ bb,
    const float* __restrict__ gam, const float* __restrict__ bet,
    const float* __restrict__ mu, const float* __restrict__ var,
    float* __restrict__ emout, int emstride, int nrows)
{
    __shared__ float h1[16][HH + 4];   // +4 pad: conflict-free column reads
    const int lane  = threadIdx.x & 31;
    const int wv    = threadIdx.x >> 5;
    const int mloc  = lane & 15;
    const int khalf = (lane >> 4) << 1;          // 0 or 2
    const int base  = blockIdx.x << 4;
    int node = base + mloc;
    if (node >= nrows) node = nrows - 1;          // clamp loads; stores guarded below
    const int col   = (wv << 4) + mloc;
    const float onepe = 1.0f + *epsp;

    // ---- GEMM1: A = (1+eps)*x + agg, B = wa[K x 64] ----
    v8f acc = {};
    for (int k = 0; k < K; k += 4) {
        int kk = k + khalf;
        v2f a, b;
        a[0] = onepe * xin[node * xstride + kk]     + agg[node * K + kk];
        a[1] = onepe * xin[node * xstride + kk + 1] + agg[node * K + kk + 1];
        b[0] = wa[kk * HH + col];
        b[1] = wa[(kk + 1) * HH + col];
        acc = __builtin_amdgcn_wmma_f32_16x16x4_f32(false, a, false, b,
                                                    (short)0, acc, false, false);
    }
    const int r0 = (lane < 16) ? 0 : 8;
    const float bav = ba[col];
#pragma unroll
    for (int r = 0; r < 8; ++r)
        h1[r + r0][col] = fmaxf(acc[r] + bav, 0.0f);
    __syncthreads();

    // ---- GEMM2: A = h1 (LDS), B = wb[64 x 64] ----
    v8f acc2 = {};
#pragma unroll 4
    for (int k = 0; k < HH; k += 4) {
        int kk = k + khalf;
        v2f a, b;
        a[0] = h1[mloc][kk];
        a[1] = h1[mloc][kk + 1];
        b[0] = wb[kk * HH + col];
        b[1] = wb[(kk + 1) * HH + col];
        acc2 = __builtin_amdgcn_wmma_f32_16x16x4_f32(false, a, false, b,
                                                     (short)0, acc2, false, false);
    }
    const float scale = gam[col] * rsqrtf(var[col] + 1e-5f);
    const float shift = bet[col] - mu[col] * scale;
    const float bbv   = bb[col];
#pragma unroll
    for (int r = 0; r < 8; ++r) {
        int n = base + r + r0;
        if (n < nrows)
            emout[n * emstride + col] = fmaxf(acc2[r] + bbv, 0.0f) * scale + shift;
    }
}

__global__ __launch_bounds__(256) void pool_sum(const float* __restrict__ em,
                                                const int* __restrict__ batch,
                                                float* __restrict__ psum, int total) {
    int i = blockIdx.x * blockDim.x + threadIdx.x;
    if (i >= total) return;
    int n = i / (LL * HH);
    int f = i - n * (LL * HH);
    atomicAdd(&psum[batch[n] * (LL * HH) + f], em[i]);
}

__global__ __launch_bounds__(256) void pool_cnt(const int* __restrict__ batch,
                                                float* __restrict__ pcnt, int n) {
    int i = blockIdx.x * blockDim.x + threadIdx.x;
    if (i < n) atomicAdd(&pcnt[batch[i]], 1.0f);
}

// pooled[512 x 320] @ lin1_w[320 x 64] + b, relu -> z[512 x 64]; fp32 WMMA
__global__ __launch_bounds__(128) void head1(const float* __restrict__ psum,
                                             const float* __restrict__ pcnt,
                                             const float* __restrict__ w1,
                                             const float* __restrict__ b1,
                                             float* __restrict__ z)
{
    const int lane  = threadIdx.x & 31;
    const int wv    = threadIdx.x >> 5;
    const int mloc  = lane & 15;
    const int khalf = (lane >> 4) << 1;
    const int base  = blockIdx.x << 4;           // 32 blocks * 16 = 512 rows exactly
    const int row   = base + mloc;
    const int col   = (wv << 4) + mloc;
    const float inv = 1.0f / fmaxf(pcnt[row], 1.0f);
    const int K = LL * HH;                        // 320

    v8f acc = {};
    for (int k = 0; k < K; k += 4) {
        int kk = k + khalf;
        v2f a, b;
        a[0] = psum[row * K + kk]     * inv;
        a[1] = psum[row * K + kk + 1] * inv;
        b[0] = w1[kk * HH + col];
        b[1] = w1[(kk + 1) * HH + col];
        acc = __builtin_amdgcn_wmma_f32_16x16x4_f32(false, a, false, b,
                                                    (short)0, acc, false, false);
    }
    const int r0 = (lane < 16) ? 0 : 8;
    const float b1v = b1[col];
#pragma unroll
    for (int r = 0; r < 8; ++r)
        z[(base + r + r0) * HH + col] = fmaxf(acc[r] + b1v, 0.0f);
}

// z[g,64] @ lin2_w[64,10] + b2, log_softmax -> out[g,10]; one wave per graph
__global__ __launch_bounds__(32) void head2(const float* __restrict__ z,
                                            const float* __restrict__ w2,
                                            const float* __restrict__ b2,
                                            float* __restrict__ out)
{
    __shared__ float lg[CC];
    const int g = blockIdx.x;
    const int t = threadIdx.x;
    if (t < CC) {
        float s = b2[t];
#pragma unroll 8
        for (int k = 0; k < HH; ++k) s += z[g * HH + k] * w2[k * CC + t];
        lg[t] = s;
    }
    __syncthreads();
    if (t == 0) {
        float mx = lg[0];
        for (int c = 1; c < CC; ++c) mx = fmaxf(mx, lg[c]);
        float se = 0.0f;
        for (int c = 0; c < CC; ++c) se += expf(lg[c] - mx);
        float lse = mx + logf(se);
        for (int c = 0; c < CC; ++c) out[g * CC + c] = lg[c] - lse;
    }
}

extern "C" void kernel_launch(void* const* d_in, const int* in_sizes, int n_in,
                              void* d_out, int out_size, void* d_ws, size_t ws_size,
                              hipStream_t stream) {
    const float* x    = (const float*)d_in[0];
    const int*   ei   = (const int*)d_in[1];
    const int*   bat  = (const int*)d_in[2];
    const float* eps1 = (const float*)d_in[3];
    const float* w1a  = (const float*)d_in[4];
    const float* b1a  = (const float*)d_in[5];
    const float* w1b  = (const float*)d_in[6];
    const float* b1b  = (const float*)d_in[7];
    const float* g1   = (const float*)d_in[8];
    const float* be1  = (const float*)d_in[9];
    const float* m1   = (const float*)d_in[10];
    const float* v1   = (const float*)d_in[11];
    const float* ceps = (const float*)d_in[12];
    const float* cwa  = (const float*)d_in[13];
    const float* cba  = (const float*)d_in[14];
    const float* cwb  = (const float*)d_in[15];
    const float* cbb  = (const float*)d_in[16];
    const float* cg   = (const float*)d_in[17];
    const float* cbe  = (const float*)d_in[18];
    const float* cm   = (const float*)d_in[19];
    const float* cv   = (const float*)d_in[20];
    const float* l1w  = (const float*)d_in[21];
    const float* l1b  = (const float*)d_in[22];
    const float* l2w  = (const float*)d_in[23];
    const float* l2b  = (const float*)d_in[24];

    const int* src = ei;          // edge_index[0,:]
    const int* dst = ei + Ee;     // edge_index[1,:]

    float* out = (float*)d_out;            // [G*C] log_softmax
    float* em  = out + GG * CC;            // [N, L*H] row-major, stride 320

    float* agg  = (float*)d_ws;            // N*128 floats
    float* psum = agg + Nn * FIN;          // G*320
    float* pcnt = psum + GG * LL * HH;     // G
    float* zbuf = pcnt + GG;               // G*64

    const int emstride = LL * HH;
    const int nblk = (Nn + 15) / 16;

    // ---------- layer 0 (K = 128) ----------
    {
        const int F = FIN, lf = 7;
        const int na = Nn * F;
        zero_f32<<<(na + 255) / 256, 256, 0, stream>>>(agg, na);
        const int tot = Ee * F;
        scatter_add<<<(tot + 255) / 256, 256, 0, stream>>>(x, FIN, src, dst, agg, lf, tot);
        gin_mlp<<<nblk, 128, 0, stream>>>(x, FIN, F, agg, eps1,
                                          w1a, b1a, w1b, b1b, g1, be1, m1, v1,
                                          em, emstride, Nn);
    }
    // ---------- layers 1..4 (K = 64), input is previous em slice ----------
    for (int l = 1; l < LL; ++l) {
        const int F = HH, lf = 6;
        const int na = Nn * F;
        const int j = l - 1;
        const float* xin = em + j * HH;
        zero_f32<<<(na + 255) / 256, 256, 0, stream>>>(agg, na);
        const int tot = Ee * F;
        scatter_add<<<(tot + 255) / 256, 256, 0, stream>>>(xin, emstride, src, dst, agg, lf, tot);
        gin_mlp<<<nblk, 128, 0, stream>>>(xin, emstride, F, agg, ceps + j,
                                          cwa + j * HH * HH, cba + j * HH,
                                          cwb + j * HH * HH, cbb + j * HH,
                                          cg + j * HH, cbe + j * HH,
                                          cm + j * HH, cv + j * HH,
                                          em + l * HH, emstride, Nn);
    }
    // ---------- global mean pool ----------
    zero_f32<<<(GG * LL * HH + 255) / 256, 256, 0, stream>>>(psum, GG * LL * HH);
    zero_f32<<<(GG + 255) / 256, 256, 0, stream>>>(pcnt, GG);
    pool_sum<<<(Nn * LL * HH + 255) / 256, 256, 0, stream>>>(em, bat, psum, Nn * LL * HH);
    pool_cnt<<<(Nn + 255) / 256, 256, 0, stream>>>(bat, pcnt, Nn);
    // ---------- heads ----------
    head1<<<GG / 16, 128, 0, stream>>>(psum, pcnt, l1w, l1b, zbuf);
    head2<<<GG, 32, 0, stream>>>(zbuf, l2w, l2b, out);
}